// GAT_45380624449851
// MI455X (gfx1250) — compile-verified
//
#include <hip/hip_runtime.h>
#include <math.h>

// ---------------- problem constants (match reference) ----------------
#define N_NODES   20000
#define N_EDGES   320000
#define E2        (N_EDGES + N_NODES)   // edges + self loops
#define HEADS     2
#define BGRAPH    32
#define SEQ_L     512
#define EMB       768
#define NEG_SLOPE 0.2f

typedef __attribute__((ext_vector_type(2))) float v2f;
typedef __attribute__((ext_vector_type(8))) float v8f;

// ---------------- small device helpers ----------------
__device__ __forceinline__ void atomicMaxF(float* addr, float v) {
  // IEEE-ordered float max via integer atomics (works with -inf init)
  if (v >= 0.0f) atomicMax((int*)addr, __float_as_int(v));
  else           atomicMin((unsigned int*)addr, __float_as_uint(v));
}

__device__ __forceinline__ void atomAddF(float* p, float v) {
  unsafeAtomicAdd(p, v);   // lowers to global_atomic_add_f32
}

__device__ __forceinline__ float sigmoidf(float x) {
  return 1.0f / (1.0f + expf(-x));
}

// ---------------- generic fill ----------------
__global__ void k_fill(float* __restrict__ p, float v, size_t n) {
  size_t i = (size_t)blockIdx.x * blockDim.x + threadIdx.x;
  if (i < n) p[i] = v;
}

// ---------------- WMMA f32 GEMM: C[M,Nc] = A[M,K] @ B[K,Nc] ----------------
// One block per 16-row M-strip; Nc/64 waves each own a 16x64 C tile (4 v8f accs).
// A strip (16 x K, contiguous in memory) is staged into LDS once per block via
// CDNA5 async global->LDS b128 copies (ASYNCcnt), with row stride K+4 floats so
// the 16-lane ds_load_b64 A-fragment fetch is bank-conflict free.
// V_WMMA_F32_16X16X4_F32 fragment layout:
//   A (16x4): lanes 0-15 -> K=k,k+1 ; lanes 16-31 -> K=k+2,k+3 ; M = lane&15
//   B (4x16): VGPR0 -> rows k / k+2 ; VGPR1 -> rows k+1 / k+3 ; N = lane&15
//   C/D     : elem r of lane l -> row (l>>4)*8 + r, col l&15
__global__ void k_gemm_wmma(const float* __restrict__ A, const float* __restrict__ B,
                            float* __restrict__ C, int K, int Nc) {
  extern __shared__ float lds_a[];           // [16][K+4]
  const int strideL = K + 4;
  const int tM = blockIdx.x;

  // ---- async copy of the contiguous 16xK A strip into padded LDS rows ----
  const float* Astrip = A + (size_t)tM * 16 * K;
  unsigned ldsBase = (unsigned)(size_t)(void*)lds_a;   // low 32 bits = LDS offset
  const int chunksPerRow = K >> 2;                     // float4 chunks per row
  const int nChunks = 16 * chunksPerRow;
  for (int i = threadIdx.x; i < nChunks; i += blockDim.x) {
    int row = i / chunksPerRow;
    int cc  = i - row * chunksPerRow;
    const float* g = Astrip + (size_t)row * K + cc * 4;
    unsigned l = ldsBase + (unsigned)(row * strideL + cc * 4) * 4u;
    asm volatile("global_load_async_to_lds_b128 %0, %1, off"
                 :: "v"(l), "v"(g) : "memory");
  }
  asm volatile("s_wait_asynccnt 0x0" ::: "memory");
  __syncthreads();

  // ---- per-wave 16x64 tile ----
  const int wave = threadIdx.x >> 5;         // 0 .. Nc/64-1
  const int lane = threadIdx.x & 31;
  const int tNbase = wave * 64;
  const int m  = lane & 15;                  // A row / B,C col within tile
  const int kh = lane >> 4;                  // which K-pair this half-wave owns

  v8f acc0 = {}, acc1 = {}, acc2 = {}, acc3 = {};
  const float* ldsRow = lds_a + m * strideL;

  #pragma unroll 2
  for (int k = 0; k < K; k += 4) {
    int ka = k + kh * 2;
    v2f av;
    av.x = ldsRow[ka];
    av.y = ldsRow[ka + 1];
    const float* b0 = B + (size_t)ka * Nc + tNbase + m;
    const float* b1 = b0 + Nc;
    v2f bv;
    bv.x = b0[0];  bv.y = b1[0];
    acc0 = __builtin_amdgcn_wmma_f32_16x16x4_f32(false, av, false, bv, (short)0, acc0, false, false);
    bv.x = b0[16]; bv.y = b1[16];
    acc1 = __builtin_amdgcn_wmma_f32_16x16x4_f32(false, av, false, bv, (short)0, acc1, false, false);
    bv.x = b0[32]; bv.y = b1[32];
    acc2 = __builtin_amdgcn_wmma_f32_16x16x4_f32(false, av, false, bv, (short)0, acc2, false, false);
    bv.x = b0[48]; bv.y = b1[48];
    acc3 = __builtin_amdgcn_wmma_f32_16x16x4_f32(false, av, false, bv, (short)0, acc3, false, false);
  }

  float* Cp = C + (size_t)(tM * 16 + kh * 8) * Nc + tNbase + m;
  #pragma unroll
  for (int r = 0; r < 8; ++r) {
    float* cr = Cp + (size_t)r * Nc;
    cr[0]  = acc0[r];
    cr[16] = acc1[r];
    cr[32] = acc2[r];
    cr[48] = acc3[r];
  }
}

// ---------------- per-node attention scores ----------------
// asrc[n,h] = sum_c h[n,h,c]*att_src[h,c] ; adst likewise
__global__ void k_att(const float* __restrict__ hbuf, const float* __restrict__ as_,
                      const float* __restrict__ ad_, float* __restrict__ asrc,
                      float* __restrict__ adst, int Cc) {
  int i = blockIdx.x * blockDim.x + threadIdx.x;
  if (i >= N_NODES * HEADS) return;
  int n = i / HEADS, h = i - n * HEADS;
  const float* hp = hbuf + (size_t)n * HEADS * Cc + h * Cc;
  const float* sa = as_ + h * Cc;
  const float* da = ad_ + h * Cc;
  float s = 0.f, d = 0.f;
  for (int c = 0; c < Cc; ++c) {
    float v = hp[c];
    s = fmaf(v, sa[c], s);
    d = fmaf(v, da[c], d);
  }
  asrc[i] = s;
  adst[i] = d;
}

// ---------------- edge pass 1: leaky-relu score + segment max ----------------
__global__ void k_edge_scores(const int* __restrict__ ei,
                              const float* __restrict__ asrc, const float* __restrict__ adst,
                              float* __restrict__ eval, float* __restrict__ emax) {
  int idx = blockIdx.x * blockDim.x + threadIdx.x;
  if (idx >= E2 * HEADS) return;
  int e = idx / HEADS, h = idx - e * HEADS;
  int s = (e < N_EDGES) ? ei[e]            : (e - N_EDGES);
  int d = (e < N_EDGES) ? ei[N_EDGES + e]  : (e - N_EDGES);
  float v = asrc[s * HEADS + h] + adst[d * HEADS + h];
  v = (v > 0.f) ? v : NEG_SLOPE * v;
  eval[idx] = v;
  atomicMaxF(&emax[d * HEADS + h], v);
}

// ---------------- edge pass 2: exp + segment sum ----------------
__global__ void k_edge_exp(const int* __restrict__ ei, float* __restrict__ eval,
                           const float* __restrict__ emax, float* __restrict__ den) {
  int idx = blockIdx.x * blockDim.x + threadIdx.x;
  if (idx >= E2 * HEADS) return;
  int e = idx / HEADS, h = idx - e * HEADS;
  int d = (e < N_EDGES) ? ei[N_EDGES + e] : (e - N_EDGES);
  float v = expf(eval[idx] - emax[d * HEADS + h]);
  eval[idx] = v;                       // eval buffer now holds ex
  atomAddF(&den[d * HEADS + h], v);
}

// ---------------- edge pass 3: alpha-weighted scatter-add ----------------
// one block per edge; threads sweep H*C features
__global__ void k_edge_agg(const int* __restrict__ ei, const float* __restrict__ hbuf,
                           const float* __restrict__ ex, const float* __restrict__ den,
                           float* __restrict__ agg, int Cc) {
  int e = blockIdx.x;
  int s = (e < N_EDGES) ? ei[e]           : (e - N_EDGES);
  int d = (e < N_EDGES) ? ei[N_EDGES + e] : (e - N_EDGES);
  int HC = HEADS * Cc;
  const float* hs = hbuf + (size_t)s * HC;
  float* ag = agg + (size_t)d * HC;
  for (int t = threadIdx.x; t < HC; t += blockDim.x) {
    int h = t / Cc;
    float alpha = ex[e * HEADS + h] / (den[d * HEADS + h] + 1e-16f);
    atomAddF(&ag[t], hs[t] * alpha);
  }
}

// ---------------- head mean + bias (+relu), write into packed output ----------------
__global__ void k_head_mean(const float* __restrict__ agg, const float* __restrict__ bias,
                            float* __restrict__ out, int Cc, int ldo, int colOff, int doRelu) {
  int idx = blockIdx.x * blockDim.x + threadIdx.x;
  if (idx >= N_NODES * Cc) return;
  int n = idx / Cc, c = idx - n * Cc;
  float s = 0.f;
  #pragma unroll
  for (int h = 0; h < HEADS; ++h) s += agg[(size_t)n * HEADS * Cc + h * Cc + c];
  s = s * (1.0f / HEADS) + bias[c];
  if (doRelu) s = fmaxf(s, 0.f);
  out[(size_t)n * ldo + colOff + c] = s;
}

// ---------------- graph pooling (mean & max per batch) ----------------
__global__ void k_pool(const float* __restrict__ h3, const int* __restrict__ batch,
                       float* __restrict__ psum, float* __restrict__ pmax,
                       float* __restrict__ pcnt) {
  int idx = blockIdx.x * blockDim.x + threadIdx.x;
  if (idx >= N_NODES * 128) return;
  int n = idx / 128, c = idx - n * 128;
  int b = batch[n];
  float v = h3[idx];
  atomAddF(&psum[b * 128 + c], v);
  atomicMaxF(&pmax[b * 128 + c], v);
  if (c == 0) atomAddF(&pcnt[b], 1.0f);
}

__global__ void k_pool_fin(const float* __restrict__ psum, const float* __restrict__ pmax,
                           const float* __restrict__ pcnt, float* __restrict__ g) {
  int idx = blockIdx.x * blockDim.x + threadIdx.x;
  if (idx >= BGRAPH * 128) return;
  int b = idx / 128, c = idx - b * 128;
  float cnt = pcnt[b];
  cnt = (cnt > 0.f) ? cnt : 1.f;
  g[b * 256 + c]       = psum[idx] / cnt;
  g[b * 256 + 128 + c] = pmax[idx];
}

// ---------------- tiny 2-layer MLP: relu(in@W1+b1)@W2+b2 -> [B] ----------------
__global__ void k_mlp(const float* __restrict__ in, const float* __restrict__ W1,
                      const float* __restrict__ b1, const float* __restrict__ W2,
                      const float* __restrict__ b2, float* __restrict__ out, int Din) {
  int b = threadIdx.x;
  if (b >= BGRAPH) return;
  float hsum[8];
  #pragma unroll
  for (int j = 0; j < 8; ++j) hsum[j] = b1[j];
  const float* row = in + (size_t)b * Din;
  for (int k = 0; k < Din; ++k) {
    float v = row[k];
    #pragma unroll
    for (int j = 0; j < 8; ++j) hsum[j] = fmaf(v, W1[k * 8 + j], hsum[j]);
  }
  float acc = b2[0];
  #pragma unroll
  for (int j = 0; j < 8; ++j) acc += fmaxf(hsum[j], 0.f) * W2[j];
  out[b] = acc;
}

// ---------------- mean over sequence dim of seq_hidden[B,L,EMB] ----------------
__global__ void k_seqmean(const float* __restrict__ seq, float* __restrict__ out) {
  int b  = blockIdx.x;
  int d0 = blockIdx.y * 256 + threadIdx.x;
  if (d0 >= EMB) return;
  const float* p = seq + (size_t)b * SEQ_L * EMB + d0;
  float s = 0.f;
  for (int l = 0; l < SEQ_L; ++l) s += p[(size_t)l * EMB];
  out[b * EMB + d0] = s * (1.0f / SEQ_L);
}

// ---------------- final sigmoid combine -> d_out[96] ----------------
__global__ void k_final(const float* __restrict__ gl, const float* __restrict__ lg,
                        float* __restrict__ out) {
  int b = threadIdx.x;
  if (b >= BGRAPH) return;
  out[b]              = sigmoidf(lg[b] + gl[b]);
  out[BGRAPH + b]     = sigmoidf(gl[b]);
  out[2 * BGRAPH + b] = sigmoidf(lg[b]);
}

// ---------------- host-side GAT stage orchestration ----------------
static void gat_stage(hipStream_t stream, const float* xin, int Din, int Cc,
                      const float* W, const float* as_, const float* ad_, const float* bias,
                      const int* ei,
                      float* hbuf, float* aggbuf, float* asrc, float* adst,
                      float* emax, float* den, float* exbuf,
                      float* outbuf, int ldo, int colOff, int doRelu) {
  const int HC = HEADS * Cc;                       // GEMM Nc (multiple of 64)
  const int threads = 32 * (HC / 64);              // one wave per 16x64 tile
  const size_t shmem = (size_t)16 * (Din + 4) * sizeof(float);
  k_gemm_wmma<<<N_NODES / 16, threads, shmem, stream>>>(xin, W, hbuf, Din, HC);

  k_att<<<(N_NODES * HEADS + 255) / 256, 256, 0, stream>>>(hbuf, as_, ad_, asrc, adst, Cc);

  size_t nAgg = (size_t)N_NODES * HC;
  k_fill<<<(int)((nAgg + 255) / 256), 256, 0, stream>>>(aggbuf, 0.0f, nAgg);
  k_fill<<<(N_NODES * HEADS + 255) / 256, 256, 0, stream>>>(emax, -INFINITY, (size_t)N_NODES * HEADS);
  k_fill<<<(N_NODES * HEADS + 255) / 256, 256, 0, stream>>>(den, 0.0f, (size_t)N_NODES * HEADS);

  int eThreads = E2 * HEADS;
  k_edge_scores<<<(eThreads + 255) / 256, 256, 0, stream>>>(ei, asrc, adst, exbuf, emax);
  k_edge_exp   <<<(eThreads + 255) / 256, 256, 0, stream>>>(ei, exbuf, emax, den);
  k_edge_agg   <<<E2, 256, 0, stream>>>(ei, hbuf, exbuf, den, aggbuf, Cc);

  k_head_mean<<<(N_NODES * Cc + 255) / 256, 256, 0, stream>>>(aggbuf, bias, outbuf, Cc, ldo, colOff, doRelu);
}

extern "C" void kernel_launch(void* const* d_in, const int* in_sizes, int n_in,
                              void* d_out, int out_size, void* d_ws, size_t ws_size,
                              hipStream_t stream) {
  (void)in_sizes; (void)n_in; (void)out_size; (void)ws_size;

  // ----- inputs (setup_inputs insertion order; params flattened depth-first) -----
  const float* x     = (const float*)d_in[0];
  const int*   ei    = (const int*)d_in[1];
  /* d_in[2] = edge_attr (ignored by reference math) */
  const int*   batch = (const int*)d_in[3];
  const float* seqh  = (const float*)d_in[4];

  int p = 5;
  auto nf = [&]() -> const float* { return (const float*)d_in[p++]; };
  const float *WA = nf(), *asA = nf(), *adA = nf(), *bA = nf();
  const float *WB = nf(), *asB = nf(), *adB = nf(), *bB = nf();
  const float *WC = nf(), *asC = nf(), *adC = nf(), *bC = nf();
  const float *WD = nf(), *asD = nf(), *adD = nf(), *bD = nf();
  const float *W2 = nf(), *as2 = nf(), *ad2 = nf(), *b2 = nf();
  const float *W3 = nf(), *as3 = nf(), *ad3 = nf(), *b3 = nf();
  const float *mW1 = nf(), *mb1 = nf(), *mW2 = nf(), *mb2 = nf();
  const float *nW1 = nf(), *nb1 = nf(), *nW2 = nf(), *nb2 = nf();

  // ----- workspace carve-up -----
  float* w = (float*)d_ws;
  size_t off = 0;
  auto alloc = [&](size_t n) -> float* { float* r = w + off; off += n; return r; };
  float* ws_h    = alloc((size_t)N_NODES * 512);   // transformed features (h = x@W)
  float* ws_cat  = alloc((size_t)N_NODES * 512);   // concat / stage io
  float* ws_agg  = alloc((size_t)N_NODES * 512);   // attention aggregation
  float* ws_asrc = alloc((size_t)N_NODES * HEADS);
  float* ws_adst = alloc((size_t)N_NODES * HEADS);
  float* ws_emax = alloc((size_t)N_NODES * HEADS);
  float* ws_den  = alloc((size_t)N_NODES * HEADS);
  float* ws_ex   = alloc((size_t)E2 * HEADS);
  float* ws_psum = alloc((size_t)BGRAPH * 128);
  float* ws_pmax = alloc((size_t)BGRAPH * 128);
  float* ws_pcnt = alloc((size_t)BGRAPH);
  float* ws_g    = alloc((size_t)BGRAPH * 256);
  float* ws_gl   = alloc((size_t)BGRAPH);
  float* ws_seq  = alloc((size_t)BGRAPH * EMB);
  float* ws_lg   = alloc((size_t)BGRAPH);

  // ----- layer 1: four parallel GATConv(256 -> 128), relu, concat to [N,512] -----
  gat_stage(stream, x, 256, 128, WA, asA, adA, bA, ei, ws_h, ws_agg,
            ws_asrc, ws_adst, ws_emax, ws_den, ws_ex, ws_cat, 512, 0,   1);
  gat_stage(stream, x, 256, 128, WB, asB, adB, bB, ei, ws_h, ws_agg,
            ws_asrc, ws_adst, ws_emax, ws_den, ws_ex, ws_cat, 512, 128, 1);
  gat_stage(stream, x, 256, 128, WC, asC, adC, bC, ei, ws_h, ws_agg,
            ws_asrc, ws_adst, ws_emax, ws_den, ws_ex, ws_cat, 512, 256, 1);
  gat_stage(stream, x, 256, 128, WD, asD, adD, bD, ei, ws_h, ws_agg,
            ws_asrc, ws_adst, ws_emax, ws_den, ws_ex, ws_cat, 512, 384, 1);

  // ----- layer 2: GATConv(512 -> 256) (no relu); output packed [N,256] in ws_cat -----
  gat_stage(stream, ws_cat, 512, 256, W2, as2, ad2, b2, ei, ws_h, ws_agg,
            ws_asrc, ws_adst, ws_emax, ws_den, ws_ex, ws_cat, 256, 0, 0);

  // ----- layer 3: GATConv(256 -> 128) (no relu); output packed [N,128] in ws_cat -----
  gat_stage(stream, ws_cat, 256, 128, W3, as3, ad3, b3, ei, ws_h, ws_agg,
            ws_asrc, ws_adst, ws_emax, ws_den, ws_ex, ws_cat, 128, 0, 0);

  // ----- graph pooling: mean & max per batch -> g[32,256] -----
  k_fill<<<(BGRAPH * 128 + 255) / 256, 256, 0, stream>>>(ws_psum, 0.0f, (size_t)BGRAPH * 128);
  k_fill<<<(BGRAPH * 128 + 255) / 256, 256, 0, stream>>>(ws_pmax, -INFINITY, (size_t)BGRAPH * 128);
  k_fill<<<1, 256, 0, stream>>>(ws_pcnt, 0.0f, (size_t)BGRAPH);
  k_pool<<<(N_NODES * 128 + 255) / 256, 256, 0, stream>>>(ws_cat, batch, ws_psum, ws_pmax, ws_pcnt);
  k_pool_fin<<<(BGRAPH * 128 + 255) / 256, 256, 0, stream>>>(ws_psum, ws_pmax, ws_pcnt, ws_g);

  // ----- heads -----
  k_mlp<<<1, 32, 0, stream>>>(ws_g, mW1, mb1, mW2, mb2, ws_gl, 256);           // graph_logit
  k_seqmean<<<dim3(BGRAPH, (EMB + 255) / 256), 256, 0, stream>>>(seqh, ws_seq);
  k_mlp<<<1, 32, 0, stream>>>(ws_seq, nW1, nb1, nW2, nb2, ws_lg, EMB);         // logits

  k_final<<<1, 32, 0, stream>>>(ws_gl, ws_lg, (float*)d_out);
}